// AtomAttention_79087527789120
// MI455X (gfx1250) — compile-verified
//
#include <hip/hip_runtime.h>
#include <hip/hip_bf16.h>
#include <cstddef>

// ============================================================================
// AtomAttention — algebraically collapsed implementation for MI455X (gfx1250)
//
// Key identity: the reference's einsum 'bqkh,bvdh->bqdh' factorizes into
//   (sum_k att) * (sum_v v), and sum_k softmax(...) == 1 exactly.
// Hence the output is
//   out = (sigmoid(atom_embed @ Wg + bg) * Vsum[b]) @ Wo + bo,
//   Vsum[b] = (sum_n atom_embed[b,n,:]) @ Wv
// molecular_matrix / mask / ln_* / W_bias / Wq / Wk never affect the output.
// Surviving work: ~4.3 GFLOP fp32 GEMM, ~10 MB HBM traffic (~0.5 us @23.3TB/s).
// GEMMs use V_WMMA_F32_16X16X4_F32 (native f32 -> no precision change).
// Sigmoid uses v_exp_f32 + v_rcp_f32 (TRANS ops co-execute with XDL WMMA).
// ============================================================================

typedef __attribute__((ext_vector_type(2))) float v2f;
typedef __attribute__((ext_vector_type(8))) float v8f;

#define AE_B   2
#define AE_N   2048
#define AE_D   256            // embedding dim
#define AE_DH  1024           // D*H
#define AE_ROWS (AE_B * AE_N) // 4096 total rows

__device__ __forceinline__ float fast_sigmoid(float x) {
    // 1 / (1 + e^-x):  v_exp_f32 + v_rcp_f32 (no IEEE div expansion)
    return __builtin_amdgcn_rcpf(1.0f + __expf(-x));
}

// ---------------------------------------------------------------------------
// Kernel 1: asum[b,d] = sum_n atom_embed[b,n,d]   (B x D = 512 values)
// ---------------------------------------------------------------------------
__global__ __launch_bounds__(256) void colsum_kernel(
    const float* __restrict__ atom_embed, float* __restrict__ asum)
{
    const int b = blockIdx.x;       // 0..1
    const int d = threadIdx.x;      // 0..255
    const float* p = atom_embed + (size_t)b * AE_N * AE_D + d;
    float s = 0.0f;
    #pragma unroll 8
    for (int n = 0; n < AE_N; ++n) s += p[(size_t)n * AE_D];
    asum[b * AE_D + d] = s;
}

// ---------------------------------------------------------------------------
// Kernel 2: vsum[b,j] = sum_d asum[b,d] * Wv[d,j]   (B x DH = 2048 values)
// ---------------------------------------------------------------------------
__global__ __launch_bounds__(256) void vsum_kernel(
    const float* __restrict__ asum, const float* __restrict__ Wv,
    float* __restrict__ vsum)
{
    const int idx = blockIdx.x * 256 + threadIdx.x;   // 0..2047
    const int b = idx >> 10;
    const int j = idx & (AE_DH - 1);
    float s = 0.0f;
    #pragma unroll 8
    for (int d = 0; d < AE_D; ++d)
        s += asum[b * AE_D + d] * Wv[(size_t)d * AE_DH + j];
    vsum[idx] = s;
}

// ---------------------------------------------------------------------------
// Kernel 3 (WMMA): per 16-row block r0:
//   G(16x1024)   = A(16x256) @ Wg + bg        [phase 1, 8 col-tiles/wave]
//   tmp          = sigmoid(G) * Vsum[b]       -> staged in 64KB LDS
//   out(16x256)  = tmp(16x1024) @ Wo + bo     [phase 2, 2 col-tiles/wave]
//
// V_WMMA_F32_16X16X4_F32 operand layouts (wave32, ISA 7.12.2):
//   A 16x4 : lane L, vgpr j -> A[M = L%16][K = 2*(L/16) + j]   (v2f per lane)
//   B 4x16 : lane L, vgpr j -> B[K = 2*(L/16) + j][N = L%16]   (v2f per lane)
//   C/D    : lane L, vgpr v -> C[M = v + 8*(L/16)][N = L%16]   (v8f per lane)
// ---------------------------------------------------------------------------
__global__ __launch_bounds__(256) void fused_gate_out_kernel(
    const float* __restrict__ atom_embed,  // (4096, 256) row-major
    const float* __restrict__ Wg,          // (256, 1024)
    const float* __restrict__ bg,          // (1024)
    const float* __restrict__ Wo,          // (1024, 256)
    const float* __restrict__ bo,          // (256)
    const float* __restrict__ vsum,        // (2, 1024)
    float* __restrict__ out)               // (4096, 256)
{
    __shared__ float lds_tmp[16 * AE_DH];  // 64 KB -> 5 workgroups per WGP

    const int tid   = threadIdx.x;
    const int wave  = tid >> 5;            // 0..7
    const int lane  = tid & 31;
    const int lhalf = lane >> 4;           // 0 | 1
    const int lmod  = lane & 15;           // N index / M index within tile
    const int koff  = lhalf * 2;           // K sub-offset for A/B operands

    const int r0    = blockIdx.x * 16;     // global row of this 16-row block
    const int batch = r0 >> 11;            // row / 2048
    const float* vs = vsum + batch * AE_DH;

    // Warm L2 with the weight tiles this wave will stream (gfx1250 prefetch).
    __builtin_prefetch(Wg + (size_t)koff * AE_DH + wave * 128, 0, 1);
    __builtin_prefetch(Wo + (size_t)koff * AE_D  + wave * 32,  0, 1);

    // ---------------- Phase 1: G = A @ Wg ; tmp = sigmoid(G+bg)*Vsum -------
    const float* Arow = atom_embed + (size_t)(r0 + lmod) * AE_D;
    for (int i = 0; i < 8; ++i) {
        const int n0 = (wave * 8 + i) * 16;    // output column tile base
        const int j  = n0 + lmod;              // this lane's output column
        v8f acc = {};
        #pragma unroll 8
        for (int kk = 0; kk < AE_D; kk += 4) {
            // A operand: 2 consecutive f32 of this lane's row (8B aligned)
            v2f a = *(const v2f*)(Arow + kk + koff);
            // B operand: two rows of Wg at this lane's column
            v2f bm;
            bm.x = Wg[(size_t)(kk + koff)     * AE_DH + j];
            bm.y = Wg[(size_t)(kk + koff + 1) * AE_DH + j];
            acc = __builtin_amdgcn_wmma_f32_16x16x4_f32(
                false, a, false, bm, (short)0, acc, false, false);
        }
        const float bgj = bg[j];
        const float vsj = vs[j];
        #pragma unroll
        for (int v = 0; v < 8; ++v) {
            const int m = v + 8 * lhalf;       // row within the 16-row block
            lds_tmp[m * AE_DH + j] = fast_sigmoid(acc[v] + bgj) * vsj;
        }
    }
    __syncthreads();

    // ---------------- Phase 2: out = tmp @ Wo + bo -------------------------
    for (int i = 0; i < 2; ++i) {
        const int n0 = (wave * 2 + i) * 16;
        const int j  = n0 + lmod;
        v8f acc = {};
        #pragma unroll 8
        for (int kk = 0; kk < AE_DH; kk += 4) {
            // A operand from LDS (offset even -> 8B aligned ds_load_b64)
            v2f a = *(const v2f*)(&lds_tmp[lmod * AE_DH + kk + koff]);
            v2f bm;
            bm.x = Wo[(size_t)(kk + koff)     * AE_D + j];
            bm.y = Wo[(size_t)(kk + koff + 1) * AE_D + j];
            acc = __builtin_amdgcn_wmma_f32_16x16x4_f32(
                false, a, false, bm, (short)0, acc, false, false);
        }
        const float boj = bo[j];
        #pragma unroll
        for (int v = 0; v < 8; ++v) {
            const int m = v + 8 * lhalf;
            out[(size_t)(r0 + m) * AE_D + j] = acc[v] + boj;
        }
    }
}

// ---------------------------------------------------------------------------
// Host launcher. Input order (setup_inputs):
//  0 molecular_matrix (unused — provably dead w.r.t. output)
//  1 atom_embed  2 embedding_mask (unused: softmax rows sum to 1 regardless)
//  3 ln_gamma 4 ln_beta 5 W_bias 6 Wq 7 Wk (all unused)  8 Wv  9 Wg 10 bg
// 11 Wo 12 bo
// ---------------------------------------------------------------------------
extern "C" void kernel_launch(void* const* d_in, const int* in_sizes, int n_in,
                              void* d_out, int out_size, void* d_ws, size_t ws_size,
                              hipStream_t stream)
{
    const float* atom_embed = (const float*)d_in[1];
    const float* Wv = (const float*)d_in[8];
    const float* Wg = (const float*)d_in[9];
    const float* bg = (const float*)d_in[10];
    const float* Wo = (const float*)d_in[11];
    const float* bo = (const float*)d_in[12];
    float* out = (float*)d_out;

    float* asum = (float*)d_ws;          // B*D   = 512 floats
    float* vsum = asum + AE_B * AE_D;    // B*DH  = 2048 floats

    colsum_kernel<<<AE_B, 256, 0, stream>>>(atom_embed, asum);
    vsum_kernel<<<(AE_B * AE_DH) / 256, 256, 0, stream>>>(asum, Wv, vsum);
    fused_gate_out_kernel<<<AE_ROWS / 16, 256, 0, stream>>>(
        atom_embed, Wg, bg, Wo, bo, vsum, out);
}